// CausalAttentionPooling_61160334295362
// MI455X (gfx1250) — compile-verified
//
#include <hip/hip_runtime.h>

typedef __attribute__((ext_vector_type(2))) float v2f;
typedef __attribute__((ext_vector_type(8))) float v8f;

#define B_   4
#define L_   4096
#define D_   256
#define SEG  128
#define NSEG (L_ / SEG)   // 32
#define DC   16           // columns per wave
#define NDC  (D_ / DC)    // 16

// ---------------------------------------------------------------------------
// Pass 1: column sums per 128-row segment. partials[((b*NSEG+s)*D)+d]
// One thread per (b, s, d): 128 coalesced loads (consecutive d across lanes).
// ---------------------------------------------------------------------------
__global__ __launch_bounds__(256) void seg_sums(const float* __restrict__ x,
                                                float* __restrict__ part) {
    int t = blockIdx.x * blockDim.x + threadIdx.x;   // [0, B*NSEG*D)
    int d = t % D_;
    int s = (t / D_) % NSEG;
    int b = t / (D_ * NSEG);
    const float* p = x + ((size_t)b * L_ + (size_t)s * SEG) * D_ + d;
    float acc = 0.0f;
#pragma unroll 4
    for (int r = 0; r < SEG; ++r) acc += p[(size_t)r * D_];
    part[t] = acc;
}

// ---------------------------------------------------------------------------
// Pass 2: in-place exclusive scan over the NSEG segments for each (b, d).
// One thread per (b, d); 32 serial elements, trivial cost.
// ---------------------------------------------------------------------------
__global__ __launch_bounds__(256) void seg_scan(float* __restrict__ part) {
    int t = blockIdx.x * blockDim.x + threadIdx.x;   // [0, B*D)
    int d = t % D_;
    int b = t / D_;
    float acc = 0.0f;
    for (int s = 0; s < NSEG; ++s) {
        size_t idx = ((size_t)(b * NSEG + s)) * D_ + d;
        float v = part[idx];
        part[idx] = acc;
        acc += v;
    }
}

// ---------------------------------------------------------------------------
// Pass 3: one wave per (b, segment, 16-col chunk). Intra-tile prefix sums via
// V_WMMA_F32_16X16X4_F32 with a lower-triangular ones A-matrix:
//   P = T(16x16, tri) * X(16x16)  computed as 4 rank-4 WMMA updates.
// Then out[i,c] = (P[i,c] + carry[c]) * rcp(i+1); carry += P[15,c].
//
// Layouts per CDNA5 ISA 7.12.2:
//   A 16x4 f32 (2 VGPRs): lane m=lane&15, h=lane>>4 holds A[m, 2h] and A[m, 2h+1]
//   B 4x16  f32 (2 VGPRs): lane holds B[2h, m] and B[2h+1, m]
//   C/D 16x16 f32 (8 VGPRs): VGPR r holds D[r + 8h, m]
// ---------------------------------------------------------------------------
__global__ __launch_bounds__(128) void prefix_mean(const float* __restrict__ x,
                                                   const float* __restrict__ carryIn,
                                                   float* __restrict__ out) {
    int wave = blockIdx.x * (blockDim.x >> 5) + (threadIdx.x >> 5);
    int lane = threadIdx.x & 31;
    int m = lane & 15;    // column within chunk (A: row M; B/C/D: column N)
    int h = lane >> 4;    // lane-half: selects K-half (A/B) / row-half (C/D)

    int dc = wave % NDC;
    int s  = (wave / NDC) % NSEG;
    int b  = wave / (NDC * NSEG);
    int d0 = dc * DC;

    // Lower-triangular ones A-matrix, split into four 16x4 K-chunks.
    // Chunk k covers K = 4k..4k+3; this lane holds K = 4k+2h and 4k+2h+1.
    v2f a[4];
#pragma unroll
    for (int k = 0; k < 4; ++k) {
        int k0 = 4 * k + 2 * h;
        a[k].x = (k0     <= m) ? 1.0f : 0.0f;
        a[k].y = (k0 + 1 <= m) ? 1.0f : 0.0f;
    }

    // Exclusive prefix of all earlier segments for this column.
    float carry = carryIn[((size_t)(b * NSEG + s)) * D_ + d0 + m];

    const float* xb = x   + ((size_t)b * L_ + (size_t)s * SEG) * D_ + d0 + m;
    float*       ob = out + ((size_t)b * L_ + (size_t)s * SEG) * D_ + d0 + m;

    for (int t = 0; t < SEG / 16; ++t) {
        const float* xt = xb + (size_t)t * 16 * D_;

        v8f c = {};  // accumulator starts at zero
#pragma unroll
        for (int k = 0; k < 4; ++k) {
            int j0 = 4 * k + 2 * h;          // B rows this lane supplies
            v2f bb;
            bb.x = xt[(size_t)j0 * D_];
            bb.y = xt[(size_t)(j0 + 1) * D_];
            c = __builtin_amdgcn_wmma_f32_16x16x4_f32(
                    /*neg_a=*/false, a[k],
                    /*neg_b=*/false, bb,
                    /*c_mod=*/(short)0, c,
                    /*reuse_a=*/false, /*reuse_b=*/false);
        }

        // Full tile column sum = prefix row 15 = c[7] on lanes 16..31.
        float colsum = __shfl(c[7], 16 + m, 32);

        int rowBase = s * SEG + t * 16;       // global row of tile row 0
#pragma unroll
        for (int r = 0; r < 8; ++r) {
            int g = rowBase + r + 8 * h;      // global row this VGPR holds
            // v_rcp_f32 (<=1 ulp) instead of the full IEEE div expansion:
            // divisor is an exact small integer, scale error ~1e-7 relative.
            float inv = __builtin_amdgcn_rcpf((float)(g + 1));
            float val = (c[r] + carry) * inv;
            ob[(size_t)(t * 16 + r + 8 * h) * D_] = val;
        }
        carry += colsum;
    }
}

// ---------------------------------------------------------------------------
extern "C" void kernel_launch(void* const* d_in, const int* in_sizes, int n_in,
                              void* d_out, int out_size, void* d_ws, size_t ws_size,
                              hipStream_t stream) {
    const float* x = (const float*)d_in[0];
    // d_in[1] (q) is mathematically irrelevant: softmax over a constant row
    // under the causal mask is exactly uniform -> causal prefix mean.
    float* out  = (float*)d_out;
    float* part = (float*)d_ws;   // needs B*NSEG*D*4 = 128 KB

    // Pass 1: 32768 threads -> 128 blocks x 256
    seg_sums<<<(B_ * NSEG * D_) / 256, 256, 0, stream>>>(x, part);
    // Pass 2: 1024 threads -> 4 blocks x 256
    seg_scan<<<(B_ * D_) / 256, 256, 0, stream>>>(part);
    // Pass 3: 2048 waves, 4 waves per block -> 512 blocks x 128
    prefix_mean<<<(B_ * NSEG * NDC) / 4, 128, 0, stream>>>(x, part, out);
}